// CMRL_Layer_72241349919297
// MI455X (gfx1250) — compile-verified
//
#include <hip/hip_runtime.h>
#include <math.h>

typedef __attribute__((ext_vector_type(2))) float v2f;
typedef __attribute__((ext_vector_type(8))) float v8f;
typedef __attribute__((ext_vector_type(4))) unsigned int u32x4;
typedef __attribute__((ext_vector_type(8))) int i32x8;
typedef __attribute__((ext_vector_type(4))) int i32x4;

#define CELU_ALPHA 3.0f

__device__ __forceinline__ float celu3(float x) {
    return x >= 0.0f ? x : CELU_ALPHA * (__expf(x * (1.0f / CELU_ALPHA)) - 1.0f);
}
__device__ __forceinline__ float sigm(float x) {
    return 1.0f / (1.0f + __expf(-x));
}
__device__ __forceinline__ void atomAddF(float* p, float v) {
    __hip_atomic_fetch_add(p, v, __ATOMIC_RELAXED, __HIP_MEMORY_SCOPE_AGENT);
}
// sign-aware float atomic max (amax initialized to -inf bit pattern)
__device__ __forceinline__ void atomMaxF(float* addr, float val) {
    if (val >= 0.0f) atomicMax((int*)addr, __float_as_int(val));
    else             atomicMin((unsigned int*)addr, (unsigned int)__float_as_int(val));
}

// ---------------- setup: rotors for both paths + zero wsum -------------------
__global__ void k_setup(const float* __restrict__ r_vec,
                        float* __restrict__ fr,     // (2,3,32,2)
                        float* __restrict__ wsum) { // (2,)
    int t = threadIdx.x;
    if (t < 2) wsum[t] = 0.0f;
    if (t < 64) {
        int m = t >> 5, j = t & 31;
        float ax = r_vec[(2 * m) * 64 + 2 * j],     ay = r_vec[(2 * m) * 64 + 2 * j + 1];
        float bx = r_vec[(2 * m + 1) * 64 + 2 * j], by = r_vec[(2 * m + 1) * 64 + 2 * j + 1];
        float ia = rsqrtf(ax * ax + ay * ay);
        float ib = rsqrtf(bx * bx + by * by);
        ax *= ia; ay *= ia; bx *= ib; by *= ib;
        float* f = fr + m * 192;
        f[0 * 64 + 2 * j]     = ax * bx - ay * by;   // rv_a * rv_b
        f[0 * 64 + 2 * j + 1] = ax * by + ay * bx;
        f[1 * 64 + 2 * j]     = bx;
        f[1 * 64 + 2 * j + 1] = by;
        f[2 * 64 + 2 * j]     = 1.0f;                // identity
        f[2 * 64 + 2 * j + 1] = 0.0f;
    }
}

// ---------------- init kernels ----------------------------------------------
__global__ void k_init_all(float* __restrict__ z0, float* __restrict__ z1,
                           float* __restrict__ amax, float* __restrict__ den,
                           int total, int n4) {
    int i = blockIdx.x * blockDim.x + threadIdx.x;
    if (i < total) { z0[i] = 0.0f; z1[i] = 0.0f; }
    if (i < n4)    { ((int*)amax)[i] = 0xFF800000; den[i] = 0.0f; }
}
__global__ void k_init_ad(float* __restrict__ amax, float* __restrict__ den, int n4) {
    int i = blockIdx.x * blockDim.x + threadIdx.x;
    if (i < n4) { ((int*)amax)[i] = 0xFF800000; den[i] = 0.0f; }
}

// ---------------- edge pass 1: eft, logits, segment max ----------------------
// one wave32 per edge; lane = complex pair index (H/2 = 32)
__global__ __launch_bounds__(256) void k_edge1(
    const float* __restrict__ feat,     // (N,64)
    const int*   __restrict__ inst,     // (E,3), this path
    const float* __restrict__ fr,       // (3,32,2), this path
    const float* __restrict__ attn1_w,  // (4,64)
    const float* __restrict__ attn2,    // (4,64)
    float* __restrict__ eft_out,        // (E,64)
    float* __restrict__ a_out,          // (E,4)
    float* __restrict__ amax,           // (N,4), bits pre-set to -inf
    int E) {
    int e    = (blockIdx.x * blockDim.x + threadIdx.x) >> 5;
    int lane = threadIdx.x & 31;
    if (e >= E) return;
    int i0 = inst[e * 3 + 0], i1 = inst[e * 3 + 1], i2 = inst[e * 3 + 2];
    float2 f0 = *(const float2*)(feat + (size_t)i0 * 64 + 2 * lane);
    float2 f1 = *(const float2*)(feat + (size_t)i1 * 64 + 2 * lane);
    float2 f2 = *(const float2*)(feat + (size_t)i2 * 64 + 2 * lane);
    float2 r0 = *(const float2*)(fr + 2 * lane);
    float2 r1 = *(const float2*)(fr + 64 + 2 * lane);
    // sum of rotations (p=2 rotor is identity), then mean over P=3
    float sx = (f0.x * r0.x - f0.y * r0.y) + (f1.x * r1.x - f1.y * r1.y) + f2.x;
    float sy = (f0.x * r0.y + f0.y * r0.x) + (f1.x * r1.y + f1.y * r1.x) + f2.y;
    float mex = sx * (1.0f / 3.0f), mey = sy * (1.0f / 3.0f);
    float efx = celu3(celu3(mex) * sigm(mex));
    float efy = celu3(celu3(mey) * sigm(mey));
    *(float2*)(eft_out + (size_t)e * 64 + 2 * lane) = make_float2(efx, efy);
    // attention logit partials: a1 from features[seg]=f0, a2 from eft
    float p[8];
#pragma unroll
    for (int h = 0; h < 4; ++h) {
        p[h]     = f0.x * attn1_w[h * 64 + 2 * lane] + f0.y * attn1_w[h * 64 + 2 * lane + 1];
        p[4 + h] = efx  * attn2[h * 64 + 2 * lane]   + efy  * attn2[h * 64 + 2 * lane + 1];
    }
#pragma unroll
    for (int m = 16; m >= 1; m >>= 1) {
#pragma unroll
        for (int h = 0; h < 8; ++h) p[h] += __shfl_xor(p[h], m, 32);
    }
    if (lane < 4) {
        float av = celu3(celu3(p[lane]) + p[4 + lane]);
        a_out[(size_t)e * 4 + lane] = av;
        atomMaxF(amax + (size_t)i0 * 4 + lane, av);
    }
}

// ---------------- edge pass 2: exp, den, weighted scatter --------------------
__global__ __launch_bounds__(256) void k_edge2(
    const int*   __restrict__ inst,
    const float* __restrict__ eft,
    const float* __restrict__ a_out,
    const float* __restrict__ amax,
    float* __restrict__ den,            // (N,4)
    float* __restrict__ hsum,           // (N,256) plane (head-major cols)
    int E) {
    int e    = (blockIdx.x * blockDim.x + threadIdx.x) >> 5;
    int lane = threadIdx.x & 31;
    if (e >= E) return;
    int seg = inst[e * 3];
    float exv[4];
#pragma unroll
    for (int h = 0; h < 4; ++h)
        exv[h] = __expf(a_out[(size_t)e * 4 + h] - amax[(size_t)seg * 4 + h]);
    if (lane < 4) atomAddF(den + (size_t)seg * 4 + lane, exv[lane]);
    float2 v = *(const float2*)(eft + (size_t)e * 64 + 2 * lane);
#pragma unroll
    for (int h = 0; h < 4; ++h) {
        float* dst = hsum + (size_t)seg * 256 + h * 64 + 2 * lane;
        atomAddF(dst,     exv[h] * v.x);
        atomAddF(dst + 1, exv[h] * v.y);
    }
}

// ---------------- node: normalize + celu, in place ---------------------------
__global__ void k_node(float* __restrict__ z, const float* __restrict__ den, int total) {
    int i = blockIdx.x * blockDim.x + threadIdx.x;
    if (i >= total) return;
    int n = i >> 8, h = (i & 255) >> 6;
    float d = den[(size_t)n * 4 + h];
    z[i] = (d > 0.0f) ? celu3(z[i] / d) : 0.0f;
}

// ---------------- gate MLP via V_WMMA_F32_16X16X4_F32 ------------------------
// One block = 16-row slab of one z plane; 8 waves. The A-slab (16x256 f32,
// 16 KB) is staged into LDS once per block by the Tensor Data Mover with
// 4-DWORD row padding (stride 260 f32) so ds_load_b64 A-fetches are
// bank-conflict free. Stage 1: 256->256 (2 col tiles/wave). Stage 2:
// 256->128 (1 tile/wave) fused with the fw3 dot; only the per-plane scalar
// sum survives (softmax gate).
__global__ __launch_bounds__(256) void k_mlp(
    const float* __restrict__ z0, const float* __restrict__ z1,
    const float* __restrict__ fw1, const float* __restrict__ fb1,
    const float* __restrict__ fw2, const float* __restrict__ fb2,
    const float* __restrict__ fw3,
    float* __restrict__ wsum, int blocksPerPlane) {
    __shared__ float zs[16 * 260];       // TDM-staged A slab, padded rows
    __shared__ float t1[16 * 260 + 8];
    float* wsums = t1 + 16 * 260;
    int plane = blockIdx.x / blocksPerPlane;
    int rb    = (blockIdx.x % blocksPerPlane) * 16;
    const float* zp = (plane == 0) ? z0 : z1;
    int lane = threadIdx.x & 31;
    int wave = threadIdx.x >> 5;
    int rr = lane & 15;            // A row / B,D col within tile
    int kh = (lane >> 4) * 2;      // per-lane K offset within K=4 step
    int rbase = 8 * (lane >> 4);   // D row offset

#if __has_builtin(__builtin_amdgcn_tensor_load_to_lds) && __has_builtin(__builtin_amdgcn_s_wait_tensorcnt)
    if (wave == 0) {
        // Tensor DMA: 2-D tile, 256 f32 x 16 rows, global stride 256 f32,
        // LDS padding 4 DWORDs after every 256 DWORDs (row stride 260 f32).
        unsigned int ldsOff = (unsigned int)(uintptr_t)(&zs[0]);
        unsigned long long ga = (unsigned long long)(uintptr_t)(zp + (size_t)rb * 256);
        u32x4 g0 = { 1u,                       // count=1, is_restore=0, no gather
                     ldsOff,                   // lds_addr
                     (unsigned int)ga,         // global_addr[31:0]
                     (unsigned int)((ga >> 32) & 0x01FFFFFFu) | (2u << 30) }; // type=2
        i32x8 g1 = { (2 << 16) | (1 << 20) | (7 << 22) | (3 << 25),
                     // ^ data_size=4B, pad_enable, pad_interval=256dw, pad_amount=4dw
                     (int)(256u << 16),        // tensor_dim0 = 256 (low16 in [63:48])
                     (int)(16u << 16),         // tensor_dim0 hi=0 | tensor_dim1 = 16
                     (int)(256u << 16),        // tensor_dim1 hi=0 | tile_dim0 = 256
                     16,                       // tile_dim1 = 16, tile_dim2 = 0
                     256,                      // tensor_dim0_stride low32 = 256
                     0, 0 };
        i32x4 gz4 = { 0, 0, 0, 0 };            // 2-D: groups 2/3 unused
        i32x8 gz8 = { 0, 0, 0, 0, 0, 0, 0, 0 };
        __builtin_amdgcn_tensor_load_to_lds(g0, g1, gz4, gz4, gz8, 0);
        __builtin_amdgcn_s_wait_tensorcnt(0);
    }
#else
    // Fallback: cooperative stage of the A slab into LDS
    for (int idx = threadIdx.x; idx < 16 * 256; idx += 256) {
        int r = idx >> 8, c = idx & 255;
        zs[r * 260 + c] = zp[(size_t)(rb + r) * 256 + c];
    }
#endif
    __syncthreads();

    // ---- stage 1: t1 = celu(z @ fw1^T + fb1), 16x256 ; A from LDS
    int cb0 = wave * 32, cb1 = cb0 + 16;
    v8f acc0 = {0.f, 0.f, 0.f, 0.f, 0.f, 0.f, 0.f, 0.f};
    v8f acc1 = acc0;
    const float* alds1 = zs  + rr * 260 + kh;
    const float* brow0 = fw1 + (size_t)(cb0 + rr) * 256 + kh;
    const float* brow1 = fw1 + (size_t)(cb1 + rr) * 256 + kh;
    for (int kk = 0; kk < 64; ++kk) {
        v2f a  = *(const v2f*)(alds1 + kk * 4);
        v2f b0 = *(const v2f*)(brow0 + kk * 4);
        v2f b1 = *(const v2f*)(brow1 + kk * 4);
        acc0 = __builtin_amdgcn_wmma_f32_16x16x4_f32(false, a, false, b0, (short)0, acc0, false, false);
        acc1 = __builtin_amdgcn_wmma_f32_16x16x4_f32(false, a, false, b1, (short)0, acc1, false, false);
    }
    {
        float biasA = fb1[cb0 + rr], biasB = fb1[cb1 + rr];
#pragma unroll
        for (int v = 0; v < 8; ++v) {
            t1[(rbase + v) * 260 + cb0 + rr] = celu3(acc0[v] + biasA);
            t1[(rbase + v) * 260 + cb1 + rr] = celu3(acc1[v] + biasB);
        }
    }
    __syncthreads();

    // ---- stage 2: t2 = celu(t1 @ fw2^T + fb2), 16x128, fused fw3 dot
    int cb2 = wave * 16;
    v8f acc = {0.f, 0.f, 0.f, 0.f, 0.f, 0.f, 0.f, 0.f};
    const float* alds2 = t1  + rr * 260 + kh;
    const float* brow2 = fw2 + (size_t)(cb2 + rr) * 256 + kh;
    for (int kk = 0; kk < 64; ++kk) {
        v2f a = *(const v2f*)(alds2 + kk * 4);
        v2f b = *(const v2f*)(brow2 + kk * 4);
        acc = __builtin_amdgcn_wmma_f32_16x16x4_f32(false, a, false, b, (short)0, acc, false, false);
    }
    float bias2 = fb2[cb2 + rr], w3 = fw3[cb2 + rr];
    float local = 0.0f;
#pragma unroll
    for (int v = 0; v < 8; ++v) local += celu3(acc[v] + bias2) * w3;
#pragma unroll
    for (int m = 16; m >= 1; m >>= 1) local += __shfl_xor(local, m, 32);
    if (lane == 0) wsums[wave] = local;
    __syncthreads();
    if (threadIdx.x == 0) {
        float t = 0.0f;
        for (int w = 0; w < 8; ++w) t += wsums[w];
        atomAddF(wsum + plane, t);
    }
}

// ---------------- gate softmax ----------------------------------------------
__global__ void k_beta(const float* __restrict__ wsum, float* __restrict__ beta, float invN) {
    if (threadIdx.x == 0 && blockIdx.x == 0) {
        float w0 = wsum[0] * invN, w1 = wsum[1] * invN;
        float m = fmaxf(w0, w1);
        float e0 = __expf(w0 - m), e1 = __expf(w1 - m);
        float s = e0 + e1;
        beta[0] = e0 / s;
        beta[1] = e1 / s;
    }
}

// ---------------- final blend (z0 lives in d_out, overwritten in place) -----
__global__ void k_combine(float* __restrict__ out, const float* __restrict__ z1,
                          const float* __restrict__ beta, int total) {
    int i = blockIdx.x * blockDim.x + threadIdx.x;
    if (i >= total) return;
    out[i] = beta[0] * out[i] + beta[1] * z1[i];
}

extern "C" void kernel_launch(void* const* d_in, const int* in_sizes, int n_in,
                              void* d_out, int out_size, void* d_ws, size_t ws_size,
                              hipStream_t stream) {
    const float* feat    = (const float*)d_in[0];
    const float* r_vec   = (const float*)d_in[1];
    const float* attn1_w = (const float*)d_in[2];
    const float* attn2   = (const float*)d_in[3];
    const float* fw1     = (const float*)d_in[4];
    const float* fb1     = (const float*)d_in[5];
    const float* fw2     = (const float*)d_in[6];
    const float* fb2     = (const float*)d_in[7];
    const float* fw3     = (const float*)d_in[8];
    const int*   insts   = (const int*)d_in[9];

    int N     = in_sizes[0] / 64;   // H = 64
    int E     = in_sizes[9] / 6;    // M*E*P = 2*E*3
    int total = N * 256;            // D = heads*H = 256
    int n4    = N * 4;

    float* ws = (float*)d_ws;
    size_t off = 0;
    float* z1   = ws + off; off += (size_t)total;
    float* eft  = ws + off; off += (size_t)E * 64;
    float* a    = ws + off; off += (size_t)E * 4;
    float* amax = ws + off; off += (size_t)n4;
    float* den  = ws + off; off += (size_t)n4;
    float* fr   = ws + off; off += 2 * 3 * 64;
    float* wsum = ws + off; off += 2;
    float* beta = ws + off; off += 2;
    float* z0   = (float*)d_out;    // hsum0 -> celu'd z0 -> blended output

    const int TB = 256;
    k_setup<<<1, 64, 0, stream>>>(r_vec, fr, wsum);
    k_init_all<<<(total + TB - 1) / TB, TB, 0, stream>>>(z0, z1, amax, den, total, n4);

    int eblocks = (E + 7) / 8;      // 8 wave32s (edges) per 256-thread block
    for (int m = 0; m < 2; ++m) {
        const int* inst_m = insts + (size_t)m * E * 3;
        float* zp = (m == 0) ? z0 : z1;
        if (m == 1)
            k_init_ad<<<(n4 + TB - 1) / TB, TB, 0, stream>>>(amax, den, n4);
        k_edge1<<<eblocks, TB, 0, stream>>>(feat, inst_m, fr + m * 192, attn1_w, attn2,
                                            eft, a, amax, E);
        k_edge2<<<eblocks, TB, 0, stream>>>(inst_m, eft, a, amax, den, zp, E);
        k_node<<<(total + TB - 1) / TB, TB, 0, stream>>>(zp, den, total);
    }

    int bpp = N / 16;               // N = 50000 -> 3125 slabs per plane
    k_mlp<<<2 * bpp, 256, 0, stream>>>(z0, z1, fw1, fb1, fw2, fb2, fw3, wsum, bpp);
    k_beta<<<1, 32, 0, stream>>>(wsum, beta, 1.0f / (float)N);
    k_combine<<<(total + TB - 1) / TB, TB, 0, stream>>>(z0, z1, beta, total);
}